// EmbeddingEngine_47029891891415
// MI455X (gfx1250) — compile-verified
//
#include <hip/hip_runtime.h>
#include <hip/hip_bf16.h>

typedef __attribute__((ext_vector_type(2))) float v2f;
typedef __attribute__((ext_vector_type(8))) float v8f;

namespace {
constexpr int S = 4;
constexpr int N = 65536;   // tokens per stream
constexpr int K = 64;      // input dim
constexpr int D = 256;     // embed dim
}

// One wave computes a 16-row x 256-col output slab via v_wmma_f32_16x16x4_f32.
// Block = 256 threads = 8 waves = 128 rows. Grid = S*N/128 = 2048 blocks.
__global__ __launch_bounds__(256, 2) void embed_gemm_pe_scatter(
    const float* __restrict__ tokens,   // [S,N,K]
    const float* __restrict__ W,        // [S,K,D]
    const float* __restrict__ bias,     // [S,D]
    const float* __restrict__ pe,       // [P,D]
    const int*   __restrict__ idxs,     // [S,N] flat scatter rows
    const int*   __restrict__ idxs_pe,  // [S,N] gather rows
    float*       __restrict__ out)      // [S*N, D]
{
    const int wave = threadIdx.x >> 5;
    const int lane = threadIdx.x & 31;
    const int half = lane >> 4;        // 0: lanes 0-15, 1: lanes 16-31
    const int l16  = lane & 15;
    const int kb   = half << 1;        // K sub-offset per ISA A/B f32 layout

    const int row0 = blockIdx.x * 128 + wave * 16;  // global row tile base
    const int s    = row0 >> 16;                    // row0 / N (N == 65536)

    // ---- Preload A fragments: 16 K-steps of the 16x64 token tile ----
    // ISA 32-bit A 16x4 layout: lane m=l16; v0=K(kb), v1=K(kb+1).
    const float* arow = tokens + (size_t)row0 * K + (size_t)l16 * K + kb;
    v2f a[16];
#pragma unroll
    for (int k = 0; k < 16; ++k)
        a[k] = *(const v2f*)(arow + 4 * k);

    // ---- Per-lane scatter/gather row indices for the 8 C rows ----
    // C/D layout: VGPR r holds row M = r + 8*half, col N = l16.
    int orow[8], prow[8];
#pragma unroll
    for (int r = 0; r < 8; ++r) {
        const int g = row0 + r + 8 * half;
        orow[r] = idxs[g];
        prow[r] = idxs_pe[g];
    }

    const float* Ws = W    + (size_t)s * K * D;
    const float* bs = bias + (size_t)s * D;

#pragma unroll 1
    for (int ct = 0; ct < 16; ++ct) {           // 16 column tiles of 16
        const int col = ct * 16 + l16;
        v8f c = {};
        // B f32 4x16 layout: v0 = row K=kb (cols=l16), v1 = row K=kb+1.
        const float* wp = Ws + (size_t)kb * D + col;
#pragma unroll
        for (int k = 0; k < 16; ++k) {          // K = 64 = 16 steps of 4
            v2f b;
            b.x = wp[0];
            b.y = wp[D];
            wp += 4 * D;
            // 8 args: (neg_a, A, neg_b, B, c_mod, C, reuse_a, reuse_b)
            c = __builtin_amdgcn_wmma_f32_16x16x4_f32(
                    false, a[k], false, b, (short)0, c, false, false);
        }
        // ---- Epilogue: bias + PE gather + indexed scatter store ----
        const float bv = bs[col];
#pragma unroll
        for (int r = 0; r < 8; ++r) {
            const float v = c[r] + bv + pe[(size_t)prow[r] * D + col];
            out[(size_t)orow[r] * D + col] = v;
        }
    }
}

extern "C" void kernel_launch(void* const* d_in, const int* in_sizes, int n_in,
                              void* d_out, int out_size, void* d_ws, size_t ws_size,
                              hipStream_t stream) {
    const float* tokens  = (const float*)d_in[0];
    const float* W       = (const float*)d_in[1];
    const float* bias    = (const float*)d_in[2];
    const float* pe      = (const float*)d_in[3];
    const int*   idxs    = (const int*)d_in[4];
    const int*   idxs_pe = (const int*)d_in[5];
    float*       out     = (float*)d_out;

    dim3 grid((S * N) / 128);
    dim3 block(256);
    embed_gemm_pe_scatter<<<grid, block, 0, stream>>>(
        tokens, W, bias, pe, idxs, idxs_pe, out);
}